// SmallTargetAwareLoss_16965120819239
// MI455X (gfx1250) — compile-verified
//
#include <hip/hip_runtime.h>
#include <math.h>
#include <stdint.h>

// ---------------- problem constants ----------------
#define B_SAMPLES 64
#define PIXELS    (512 * 512)          // 262144 per sample
#define TPB       256                  // 8 wave32s per block
#define BLKS      32                   // blocks per sample
#define VEC_ITERS 8                    // float4 per thread: 32*256*8*4 == PIXELS
#define THRESHOLD 0.05f
#define SMOOTH    1e-5f

// ---------------- CDNA5 async-to-LDS path (probe via __has_builtin) ----------------
#if defined(__gfx1250__) && defined(__has_builtin)
#if __has_builtin(__builtin_amdgcn_global_load_async_to_lds_b128) && \
    __has_builtin(__builtin_amdgcn_s_wait_asynccnt)
#define USE_ASYNC_LDS 1
#endif
#endif
#ifndef USE_ASYNC_LDS
#define USE_ASYNC_LDS 0
#endif

#if USE_ASYNC_LDS
typedef int v4i __attribute__((ext_vector_type(4)));
typedef __attribute__((address_space(1))) v4i gv4i;   // global int4
typedef __attribute__((address_space(3))) v4i lv4i;   // LDS int4
#define ASYNC_B128(gp, lp) \
  __builtin_amdgcn_global_load_async_to_lds_b128((gv4i*)(void*)(gp), (lv4i*)(void*)(lp), 0, 0)
#endif

// ---------------- helpers ----------------
__device__ __forceinline__ float wave_sum32(float v) {
#pragma unroll
  for (int off = 16; off > 0; off >>= 1) v += __shfl_down(v, off, 32);
  return v;
}

// per-element fused math: stable BCE * pos_weight, sigmoid, and the 4 sums
__device__ __forceinline__ void accum4(const float4 x4, const float4 t4,
                                       float& wbce, float& inter,
                                       float& psum, float& tsum) {
  const float xs[4] = {x4.x, x4.y, x4.z, x4.w};
  const float ts[4] = {t4.x, t4.y, t4.z, t4.w};
#pragma unroll
  for (int c = 0; c < 4; ++c) {
    const float x = xs[c];
    const float t = ts[c];
    const float e   = __expf(-fabsf(x));          // e = exp(-|x|) in (0,1]
    const float lg  = __logf(1.0f + e);           // log1p(e), 1+e in (1,2]
    const float bce = fmaxf(x, 0.0f) - x * t + lg;
    const float pw  = fminf(fmaf(10.0f, t, 1.0f), 20.0f);
    wbce = fmaf(bce, pw, wbce);
    const float inv = __builtin_amdgcn_rcpf(1.0f + e); // v_rcp_f32
    const float p   = (x >= 0.0f) ? inv : e * inv;     // sigmoid(x)
    psum += p;
    inter = fmaf(p, t, inter);
    tsum += t;
  }
}

// ---------------- kernel 1: per-block partial sums ----------------
__global__ __launch_bounds__(TPB) void stal_partial(
    const float4* __restrict__ pred, const float4* __restrict__ targ,
    float4* __restrict__ partials) {
  const int tid  = threadIdx.x;
  const int blk  = blockIdx.x;   // 0..BLKS-1 chunk within sample
  const int b    = blockIdx.y;   // sample
  const int base = b * (PIXELS / 4) + blk * (TPB * VEC_ITERS); // float4 units

  float wbce = 0.0f, inter = 0.0f, psum = 0.0f, tsum = 0.0f;

#if USE_ASYNC_LDS
  // Double-buffered async staging: global -> LDS, tracked by ASYNCcnt.
  __shared__ float4 sP[2][TPB];
  __shared__ float4 sT[2][TPB];
  {
    const int gidx = base + tid;
    ASYNC_B128(pred + gidx, &sP[0][tid]);
    ASYNC_B128(targ + gidx, &sT[0][tid]);
  }
#pragma unroll
  for (int it = 0; it < VEC_ITERS; ++it) {
    const int st = it & 1;
    if (it + 1 < VEC_ITERS) {
      const int gidx = base + (it + 1) * TPB + tid;
      ASYNC_B128(pred + gidx, &sP[st ^ 1][tid]);
      ASYNC_B128(targ + gidx, &sT[st ^ 1][tid]);
      __builtin_amdgcn_s_wait_asynccnt(2);   // oldest 2 (current stage) done
    } else {
      __builtin_amdgcn_s_wait_asynccnt(0);
    }
    __asm__ __volatile__("" ::: "memory");   // no LDS reads hoisted above wait
    const float4 x4 = sP[st][tid];           // lane reads only its own slot
    const float4 t4 = sT[st][tid];
    accum4(x4, t4, wbce, inter, psum, tsum);
    __asm__ __volatile__("" ::: "memory");   // reads done before next overwrite
  }
#else
  // Fallback: direct b128 streaming + CDNA5 global_prefetch_b8
#pragma unroll
  for (int it = 0; it < VEC_ITERS; ++it) {
    const int gidx = base + it * TPB + tid;
    if (it + 1 < VEC_ITERS) {
      __builtin_prefetch(pred + gidx + TPB, 0, 3);
      __builtin_prefetch(targ + gidx + TPB, 0, 3);
    }
    const float4 x4 = pred[gidx];
    const float4 t4 = targ[gidx];
    accum4(x4, t4, wbce, inter, psum, tsum);
  }
#endif

  // wave32 reduction, then cross-wave via LDS
  wbce  = wave_sum32(wbce);
  inter = wave_sum32(inter);
  psum  = wave_sum32(psum);
  tsum  = wave_sum32(tsum);

  __shared__ float4 wpart[TPB / 32];
  const int wave = tid >> 5, lane = tid & 31;
  if (lane == 0) wpart[wave] = make_float4(wbce, inter, psum, tsum);
  __syncthreads();
  if (tid == 0) {
    float4 a = wpart[0];
#pragma unroll
    for (int w = 1; w < TPB / 32; ++w) {
      const float4 v = wpart[w];
      a.x += v.x; a.y += v.y; a.z += v.z; a.w += v.w;
    }
    partials[b * BLKS + blk] = a;
  }
}

// ---------------- kernel 2: per-sample loss + validity ----------------
__global__ __launch_bounds__(32) void stal_sample(
    const float4* __restrict__ partials, float2* __restrict__ sres) {
  const int b = blockIdx.x;
  const int t = threadIdx.x;                 // 32 threads == BLKS partials
  const float4 v = partials[b * BLKS + t];
  const float wbce  = wave_sum32(v.x);
  const float inter = wave_sum32(v.y);
  const float psum  = wave_sum32(v.z);
  const float tsum  = wave_sum32(v.w);
  if (t == 0) {
    const float invP = 1.0f / (float)PIXELS;
    const float wbce_mean = wbce * invP;
    const float dice = 1.0f - (2.0f * inter + SMOOTH) / (psum + tsum + SMOOTH);
    const float loss = 0.6f * wbce_mean + 0.4f * dice;
    const bool valid = (tsum * invP) < THRESHOLD;
    sres[b] = valid ? make_float2(loss, 1.0f) : make_float2(0.0f, 0.0f);
  }
}

// ---------------- kernel 3: final masked average ----------------
__global__ __launch_bounds__(32) void stal_final(
    const float2* __restrict__ sres, float* __restrict__ out) {
  const int t = threadIdx.x;
  const float2 a = sres[t];
  const float2 c = sres[t + 32];
  float tot = a.x + c.x;
  float nv  = a.y + c.y;
  tot = wave_sum32(tot);
  nv  = wave_sum32(nv);
  if (t == 0) out[0] = (nv > 0.0f) ? (tot / fmaxf(nv, 1.0f)) : 0.0f;
}

// ---------------- entry ----------------
extern "C" void kernel_launch(void* const* d_in, const int* in_sizes, int n_in,
                              void* d_out, int out_size, void* d_ws, size_t ws_size,
                              hipStream_t stream) {
  (void)in_sizes; (void)n_in; (void)out_size; (void)ws_size;
  const float4* pred = (const float4*)d_in[0];
  const float4* targ = (const float4*)d_in[1];

  float4* partials = (float4*)d_ws;                                   // 2048 * 16B
  float2* sres = (float2*)((char*)d_ws + B_SAMPLES * BLKS * sizeof(float4));

  stal_partial<<<dim3(BLKS, B_SAMPLES), TPB, 0, stream>>>(pred, targ, partials);
  stal_sample<<<B_SAMPLES, 32, 0, stream>>>(partials, sres);
  stal_final<<<1, 32, 0, stream>>>(sres, (float*)d_out);
}